// DMGI_FULL_EMBED_73624329388260
// MI455X (gfx1250) — compile-verified
//
#include <hip/hip_runtime.h>
#include <hip/hip_bf16.h>

typedef float v2f __attribute__((ext_vector_type(2)));
typedef float v8f __attribute__((ext_vector_type(8)));

#define D 128          // feature dim (in == out == 128)
#define MROWS 64       // rows per GEMM block
#define KCHUNK 64      // K chunk held in LDS
#define APAD 4         // LDS padding (floats) -> conflict-free b64 reads

__device__ __forceinline__ void atomic_add_f32(float* p, float v) {
    // native global_atomic_add_f32 (no-return, STOREcnt) on gfx1250
    unsafeAtomicAdd(p, v);
}

// ---------------------------------------------------------------------------
// Utility kernels
// ---------------------------------------------------------------------------
__global__ void zero_f32_kernel(float* __restrict__ p, long n) {
    long stride = (long)gridDim.x * blockDim.x;
    for (long i = (long)blockIdx.x * blockDim.x + threadIdx.x; i < n; i += stride)
        p[i] = 0.0f;
}

__global__ void fill_f32_kernel(float* __restrict__ p, int n, float v) {
    int i = blockIdx.x * blockDim.x + threadIdx.x;
    if (i < n) p[i] = v;
}

// deg[dst[e]] += 1  (deg pre-filled with 1.0 for the self loop)
__global__ void deg_kernel(const int* __restrict__ dst, float* __restrict__ deg, int E) {
    int e = blockIdx.x * blockDim.x + threadIdx.x;
    if (e < E) atomic_add_f32(&deg[dst[e]], 1.0f);
}

// in-place deg -> rsqrt(max(deg,1))
__global__ void rsqrt_kernel(float* __restrict__ deg, int n) {
    int i = blockIdx.x * blockDim.x + threadIdx.x;
    if (i < n) deg[i] = rsqrtf(fmaxf(deg[i], 1.0f));
}

// ---------------------------------------------------------------------------
// GEMM: xw[M x 128] = A @ W, A[r] = x[srow] * mask[srow], srow = perm?perm[r]:r
// fp32 WMMA 16x16x4; block = 256 threads (8 waves) -> 64 rows x 128 cols tile
// ---------------------------------------------------------------------------
__global__ __launch_bounds__(256)
void gemm_xw_kernel(const float* __restrict__ x, const float* __restrict__ mask,
                    const int* __restrict__ perm, const float* __restrict__ W,
                    float* __restrict__ xw, int M) {
    __shared__ float lA[MROWS][KCHUNK + APAD];     // A chunk (row-major)
    __shared__ float lBT[D][KCHUNK + APAD];        // W chunk, transposed: lBT[n][k]

    const int tid  = threadIdx.x;
    const int lane = tid & 31;
    const int wave = tid >> 5;                     // N-tile index 0..7
    const int brow = blockIdx.x * MROWS;

    v8f c[4] = {};                                 // 4 M-tiles of 16x16 f32 accum

    const int koff = 2 * (lane >> 4);              // per-lane K sub-offset (0 or 2)
    const int frow = lane & 15;                    // fragment row (A: M, B: N)

    for (int kc = 0; kc < 2; ++kc) {
        if (kc) __syncthreads();                   // protect LDS reuse
        // ---- stage A chunk: 64 rows x 64 k (fused mask, optional perm) ----
        for (int i = tid; i < MROWS * KCHUNK; i += 256) {
            int r = i >> 6, k = i & 63;
            int grow = brow + r;
            float v = 0.0f;
            if (grow < M) {
                int srow = perm ? perm[grow] : grow;
                size_t off = (size_t)srow * D + (size_t)(kc * KCHUNK + k);
                v = x[off] * mask[off];
            }
            lA[r][k] = v;
        }
        // ---- stage W^T chunk: 64 k x 128 n -> lBT[n][k] ----
        for (int i = tid; i < KCHUNK * D; i += 256) {
            int k = i >> 7, n = i & 127;
            lBT[n][k] = W[(size_t)(kc * KCHUNK + k) * D + n];
        }
        __syncthreads();

        const float* bp = &lBT[wave * 16 + frow][0];
#pragma unroll
        for (int mt = 0; mt < 4; ++mt) {
            const float* ap = &lA[mt * 16 + frow][0];
#pragma unroll
            for (int ks = 0; ks < KCHUNK / 4; ++ks) {
                v2f a = *(const v2f*)(ap + ks * 4 + koff);
                v2f b = *(const v2f*)(bp + ks * 4 + koff);
                c[mt] = __builtin_amdgcn_wmma_f32_16x16x4_f32(
                    false, a, false, b, (short)0, c[mt], false, false);
            }
        }
    }

    // ---- store: D layout: lane = n + 16*(m/8), vgpr v -> m = 8*(lane/16)+v ----
    const int col   = wave * 16 + frow;
    const int rbase = brow + 8 * (lane >> 4);
#pragma unroll
    for (int mt = 0; mt < 4; ++mt) {
#pragma unroll
        for (int v = 0; v < 8; ++v) {
            int row = rbase + mt * 16 + v;
            if (row < M) xw[(size_t)row * D + col] = c[mt][v];
        }
    }
}

// ---------------------------------------------------------------------------
// Aggregation: one wave per (edge | self-loop); 4 floats/lane, f32 atomics
// out[dst] += xw[src] * dinv[src]*dinv[dst]
// ---------------------------------------------------------------------------
__global__ __launch_bounds__(256)
void agg_kernel(const float* __restrict__ xw, const int* __restrict__ src,
                const int* __restrict__ dst, const float* __restrict__ dinv,
                float* __restrict__ out, int E, int N) {
    int item = blockIdx.x * 8 + (threadIdx.x >> 5);
    int lane = threadIdx.x & 31;
    if (item >= E + N) return;
    int s, d;
    if (item < E) { s = src[item]; d = dst[item]; }
    else          { s = item - E; d = s; }        // self-loop
    float norm = dinv[s] * dinv[d];
    const float4* xs = (const float4*)(xw + (size_t)s * D);
    float4 v = xs[lane];
    float* o = out + (size_t)d * D + lane * 4;
    atomic_add_f32(o + 0, v.x * norm);
    atomic_add_f32(o + 1, v.y * norm);
    atomic_add_f32(o + 2, v.z * norm);
    atomic_add_f32(o + 3, v.w * norm);
}

// ---------------------------------------------------------------------------
// Epilogue: out = relu(out + b); optional summary[col] += sum(out[:,col]) / M
// block = 128 threads (one per column), each block covers `rows` rows
// ---------------------------------------------------------------------------
__global__ __launch_bounds__(128)
void relu_bias_kernel(float* __restrict__ out, const float* __restrict__ b,
                      float* summary, int M, int rows) {
    int col = threadIdx.x;
    float bias = b[col];
    int r0 = blockIdx.x * rows;
    int r1 = r0 + rows; if (r1 > M) r1 = M;
    float acc = 0.0f;
    for (int r = r0; r < r1; ++r) {
        size_t idx = (size_t)r * D + col;
        float v = fmaxf(out[idx] + bias, 0.0f);
        out[idx] = v;
        acc += v;
    }
    if (summary) atomic_add_f32(&summary[col], acc * (1.0f / (float)M));
}

// ---------------------------------------------------------------------------
extern "C" void kernel_launch(void* const* d_in, const int* in_sizes, int n_in,
                              void* d_out, int out_size, void* d_ws, size_t ws_size,
                              hipStream_t stream) {
    const float* x    = (const float*)d_in[0];
    const float* W1   = (const float*)d_in[1];
    const float* b1   = (const float*)d_in[2];
    const float* W2   = (const float*)d_in[3];
    const float* b2   = (const float*)d_in[4];
    const float* mp1  = (const float*)d_in[5];
    const float* mn1  = (const float*)d_in[6];
    const float* mp2  = (const float*)d_in[7];
    const float* mn2  = (const float*)d_in[8];
    const int*   ei1  = (const int*)d_in[9];
    const int*   ei2  = (const int*)d_in[10];
    const int*   perm1 = (const int*)d_in[11];
    const int*   perm2 = (const int*)d_in[12];

    const int N = in_sizes[0] / D;
    const int E = in_sizes[9] / 2;
    const int* src1 = ei1;      const int* dst1 = ei1 + E;
    const int* src2 = ei2;      const int* dst2 = ei2 + E;

    float* out = (float*)d_out;
    const size_t ND = (size_t)N * D;
    float* p1 = out;
    float* n1 = out + ND;
    float* s1 = out + 2 * ND;
    float* p2 = s1 + D;
    float* n2 = p2 + ND;
    // s2 = n2 + ND (zeroed with the rest; filled by pos-branch epilogue)
    float* s2 = n2 + ND;

    float* ws    = (float*)d_ws;
    float* xw    = ws;                 // N*128 floats scratch (reused 4x)
    float* dinv1 = ws + ND;            // N floats (deg -> rsqrt in place)
    float* dinv2 = dinv1 + N;          // N floats

    // ---- init: zero whole output, degrees = 1.0 (self loop) ----
    zero_f32_kernel<<<2048, 256, 0, stream>>>(out, (long)out_size);
    fill_f32_kernel<<<(2 * N + 255) / 256, 256, 0, stream>>>(dinv1, 2 * N, 1.0f);
    deg_kernel<<<(E + 255) / 256, 256, 0, stream>>>(dst1, dinv1, E);
    deg_kernel<<<(E + 255) / 256, 256, 0, stream>>>(dst2, dinv2, E);
    rsqrt_kernel<<<(2 * N + 255) / 256, 256, 0, stream>>>(dinv1, 2 * N);

    const int gemmGrid = (N + MROWS - 1) / MROWS;
    const int aggGrid  = (E + N + 7) / 8;
    const int reluRows = 128;
    const int reluGrid = (N + reluRows - 1) / reluRows;

    // ---- branch 1, positive ----
    gemm_xw_kernel<<<gemmGrid, 256, 0, stream>>>(x, mp1, nullptr, W1, xw, N);
    agg_kernel<<<aggGrid, 256, 0, stream>>>(xw, src1, dst1, dinv1, p1, E, N);
    relu_bias_kernel<<<reluGrid, 128, 0, stream>>>(p1, b1, s1, N, reluRows);
    // ---- branch 1, negative (permuted rows) ----
    gemm_xw_kernel<<<gemmGrid, 256, 0, stream>>>(x, mn1, perm1, W1, xw, N);
    agg_kernel<<<aggGrid, 256, 0, stream>>>(xw, src1, dst1, dinv1, n1, E, N);
    relu_bias_kernel<<<reluGrid, 128, 0, stream>>>(n1, b1, nullptr, N, reluRows);
    // ---- branch 2, positive ----
    gemm_xw_kernel<<<gemmGrid, 256, 0, stream>>>(x, mp2, nullptr, W2, xw, N);
    agg_kernel<<<aggGrid, 256, 0, stream>>>(xw, src2, dst2, dinv2, p2, E, N);
    relu_bias_kernel<<<reluGrid, 128, 0, stream>>>(p2, b2, s2, N, reluRows);
    // ---- branch 2, negative ----
    gemm_xw_kernel<<<gemmGrid, 256, 0, stream>>>(x, mn2, perm2, W2, xw, N);
    agg_kernel<<<aggGrid, 256, 0, stream>>>(xw, src2, dst2, dinv2, n2, E, N);
    relu_bias_kernel<<<reluGrid, 128, 0, stream>>>(n2, b2, nullptr, N, reluRows);
}